// Branch_89515708383977
// MI455X (gfx1250) — compile-verified
//
#include <hip/hip_runtime.h>
#include <hip/hip_bf16.h>

typedef __attribute__((ext_vector_type(2))) float v2f;
typedef __attribute__((ext_vector_type(8))) float v8f;

// ---------------------------------------------------------------------------
// Problem constants (fixed by setup_inputs)
// ---------------------------------------------------------------------------
#define BATCH   8
#define CIN1    4
#define CMID    12
#define COUT2   36
// conv1 spatial
#define T0 20
#define D0 24
#define H0 24
#define W0 24
#define T1 18
#define D1 22
#define H1 22
#define W1d 22
// conv2 spatial
#define T2 16
#define D2 20
#define H2 20
#define W2d 20

#define SP0   (T0*D0*H0*W0)          // 276480  input spatial per (b,c)
#define NX    (BATCH*CIN1*SP0)       // 8,847,360 x elements
#define P1    (T1*D1*H1*W1d)         // 191,664 conv1 out positions / batch
#define NH    (BATCH*P1*CMID)        // 18,399,744 h elements
#define P2    (T2*D2*H2*W2d)         // 128,000 conv2 out positions / batch
#define CNT_PER_CH ((float)(BATCH*P1))   // 1,533,312 elements per channel
#define NBLK_STATS 512

// ---------------------------------------------------------------------------
// x [B][C][T][D][H][W] -> xcl [B][T][D][H][W][C]   (channel-last)
// ---------------------------------------------------------------------------
__global__ __launch_bounds__(256) void transpose_cl(const float* __restrict__ x,
                                                    float* __restrict__ xcl) {
  int i = blockIdx.x * 256 + threadIdx.x;
  if (i >= NX) return;
  int s = i % SP0;
  int c = (i / SP0) & (CIN1 - 1);
  int b = i / (CIN1 * SP0);
  xcl[((long)b * SP0 + s) * CIN1 + c] = x[i];
}

// ---------------------------------------------------------------------------
// Pre-shuffle weights into WMMA A-fragment lane order:
//   wprep[mtile][tap][kstep][lane][2]
// lane<16  holds A[m=lane][k=4j+0], A[m][4j+1]
// lane>=16 holds A[m=lane-16][k=4j+2], A[m][4j+3]     (ISA 16x4 f32 A layout)
// A[m][k] = W[m][k][it][id][ih][iw] (k = input channel), zero padded.
// ---------------------------------------------------------------------------
__global__ __launch_bounds__(256) void prep_weights(const float* __restrict__ w,
                                                    float* __restrict__ wprep,
                                                    int COUT, int CIN, int MTILES,
                                                    int KSTEPS) {
  int total = MTILES * 81 * KSTEPS * 64;
  for (int i = blockIdx.x * blockDim.x + threadIdx.x; i < total;
       i += blockDim.x * gridDim.x) {
    int v    = i & 1;
    int lane = (i >> 1) & 31;
    int rest = i >> 6;
    int j    = rest % KSTEPS; rest /= KSTEPS;
    int tap  = rest % 81;
    int mt   = rest / 81;
    int m = mt * 16 + (lane & 15);
    int k = 4 * j + ((lane >> 4) << 1) + v;
    float val = 0.f;
    if (m < COUT && k < CIN) {
      int iw = tap % 3, ih = (tap / 3) % 3, id = (tap / 9) % 3, it = tap / 27;
      val = w[((((m * CIN + k) * 3 + it) * 3 + id) * 3 + ih) * 3 + iw];
    }
    wprep[i] = val;
  }
}

// ---------------------------------------------------------------------------
// Implicit-GEMM 4D conv via V_WMMA_F32_16X16X4_F32.
//   One wave computes one 16(co) x 16(pos) output tile; fully unrolled flat
//   step stream with compile-time immediate offsets. The per-lane LDS offset
//   for A fragments is routed through an empty asm inside the tile loop:
//   loads remain plain ds_load_b64 (proper DScnt tracking, schedulable within
//   the tile body) but can no longer be LICM-hoisted across tiles -> short
//   live ranges, low VGPR pressure, high occupancy. Two independent
//   accumulator chains (even/odd steps) break the serial C-dependency.
// ---------------------------------------------------------------------------
template <int CIN, int COUT, int KSTEPS, bool CHLAST_OUT,
          int TIN, int DIN, int HIN, int WIN,
          int TOUT, int DOUT, int HOUT, int WOUT>
__global__ __launch_bounds__(256) void conv4d_wmma(const float* __restrict__ xcl,
                                                   const float* __restrict__ wprep,
                                                   const float* __restrict__ bias,
                                                   float* __restrict__ out) {
  constexpr int TAPS  = 81;
  constexpr int STEPS = TAPS * KSTEPS;
  constexpr int P     = TOUT * DOUT * HOUT * WOUT;
  constexpr int TPB   = P / 16;                // divides exactly for both convs
  __shared__ float sA[STEPS * 64];
  __shared__ float sBias[16];

  const int mtile = blockIdx.y;
  const float* wsrc = wprep + (long)mtile * (STEPS * 64);
  for (int i = threadIdx.x; i < STEPS * 64; i += blockDim.x) sA[i] = wsrc[i];
  if (threadIdx.x < 16) {
    int co = mtile * 16 + threadIdx.x;
    sBias[threadIdx.x] = (co < COUT) ? bias[co] : 0.f;
  }
  __syncthreads();

  const int lane = threadIdx.x & 31;
  const int wave = threadIdx.x >> 5;
  const int hi   = lane >> 4;        // 0: lanes 0-15, 1: lanes 16-31
  const int nIdx = lane & 15;        // N (position) within tile
  const int kOff = hi * 2;           // B rows {0,1} vs {2,3} of each 4-row slab
  const int laneOff = lane * 2;      // per-lane A offset (floats)

  const int totalTiles = BATCH * TPB;
  const int nWaves = (blockDim.x >> 5) * gridDim.x;

  for (int tile = blockIdx.x * (blockDim.x >> 5) + wave; tile < totalTiles;
       tile += nWaves) {
    int b = tile / TPB;
    int local = (tile - b * TPB) * 16 + nIdx;
    int tmp = local;
    int ww = tmp % WOUT; tmp /= WOUT;
    int hh = tmp % HOUT; tmp /= HOUT;
    int dd = tmp % DOUT;
    int tt = tmp / DOUT;
    const float* base = xcl +
        (long)CIN * (((((long)b * TIN + tt) * DIN + dd) * HIN + hh) * WIN + ww) +
        kOff;

    // opaque copy of the A lane-offset: defined inside the tile loop so the
    // sA loads below cannot be hoisted out of the loop (keeps VGPRs low).
    int lo = laneOff;
    asm volatile("" : "+v"(lo));

    v8f acc0 = {};
    v8f acc1 = {};
#pragma unroll
    for (int it = 0; it < 3; ++it)
#pragma unroll
      for (int id = 0; id < 3; ++id)
#pragma unroll
        for (int ih = 0; ih < 3; ++ih)
#pragma unroll
          for (int iw = 0; iw < 3; ++iw) {
            const int tap  = ((it * 3 + id) * 3 + ih) * 3 + iw;
            const int toff = CIN * (((it * DIN + id) * HIN + ih) * WIN + iw);
#pragma unroll
            for (int j = 0; j < KSTEPS; ++j) {
              const int s = tap * KSTEPS + j;
              v2f a  = *(const v2f*)&sA[s * 64 + lo];           // ds_load_b64
              v2f bb = *(const v2f*)(base + toff + 4 * j);      // global b64
              if (s & 1)
                acc1 = __builtin_amdgcn_wmma_f32_16x16x4_f32(
                    false, a, false, bb, (short)0, acc1, false, false);
              else
                acc0 = __builtin_amdgcn_wmma_f32_16x16x4_f32(
                    false, a, false, bb, (short)0, acc0, false, false);
            }
          }

    // epilogue: merge chains + bias + ReLU + store (vgpr r -> row r / r+8)
#pragma unroll
    for (int r = 0; r < 8; ++r) {
      int m  = r + hi * 8;
      int co = mtile * 16 + m;
      float v = acc0[r] + acc1[r] + sBias[m];
      v = v > 0.f ? v : 0.f;
      if (co < COUT) {
        if (CHLAST_OUT)
          out[((long)b * P + local) * COUT + co] = v;        // h channel-last
        else
          out[((long)b * COUT + co) * P + local] = v;        // reference layout
      }
    }
  }
}

// ---------------------------------------------------------------------------
// Per-channel sum / sumsq over channel-last h. blockDim=384 (=12*32) so each
// thread owns a fixed channel; deterministic two-level reduction (no atomics).
// ---------------------------------------------------------------------------
__global__ __launch_bounds__(384) void stats_kernel(const float* __restrict__ h,
                                                    float* __restrict__ partials) {
  const int c = threadIdx.x % 12;
  float s = 0.f, ss = 0.f;
  const long stride = (long)384 * gridDim.x;   // multiple of 12
  for (long i = (long)blockIdx.x * 384 + threadIdx.x; i < (long)NH; i += stride) {
    float v = h[i];
    s += v;
    ss += v * v;
  }
  __shared__ float red[768];
  red[threadIdx.x] = s;
  red[384 + threadIdx.x] = ss;
  __syncthreads();
  if (threadIdx.x < 12) {
    float ts = 0.f, tss = 0.f;
    for (int g = 0; g < 32; ++g) {
      ts  += red[g * 12 + c];
      tss += red[384 + g * 12 + c];
    }
    partials[blockIdx.x * 24 + c]      = ts;
    partials[blockIdx.x * 24 + 12 + c] = tss;
  }
}

__global__ void finalize_stats(const float* __restrict__ partials,
                               const float* __restrict__ gamma,
                               const float* __restrict__ beta,
                               float* __restrict__ scaleshift) {
  int c = threadIdx.x;
  if (c < 12) {
    float s = 0.f, ss = 0.f;
    for (int b = 0; b < NBLK_STATS; ++b) {
      s  += partials[b * 24 + c];
      ss += partials[b * 24 + 12 + c];
    }
    float mean = s / CNT_PER_CH;
    float var  = ss / CNT_PER_CH - mean * mean;   // biased, matches jnp.var
    float sc   = gamma[c] * rsqrtf(var + 1e-3f);  // EPS = 0.001
    scaleshift[c]      = sc;
    scaleshift[12 + c] = beta[c] - mean * sc;
  }
}

__global__ __launch_bounds__(384) void norm_kernel(float* __restrict__ h,
                                                   const float* __restrict__ scaleshift) {
  const int c = threadIdx.x % 12;
  const float sc = scaleshift[c];
  const float sh = scaleshift[12 + c];
  const long stride = (long)384 * gridDim.x;
  for (long i = (long)blockIdx.x * 384 + threadIdx.x; i < (long)NH; i += stride)
    h[i] = h[i] * sc + sh;
}

// ---------------------------------------------------------------------------
extern "C" void kernel_launch(void* const* d_in, const int* in_sizes, int n_in,
                              void* d_out, int out_size, void* d_ws, size_t ws_size,
                              hipStream_t stream) {
  const float* x     = (const float*)d_in[0];
  const float* Wc1   = (const float*)d_in[1];
  const float* b1    = (const float*)d_in[2];
  const float* gamma = (const float*)d_in[3];
  const float* beta  = (const float*)d_in[4];
  const float* Wc2   = (const float*)d_in[5];
  const float* b2    = (const float*)d_in[6];
  float* out = (float*)d_out;

  float* W = (float*)d_ws;
  float* xcl   = W;                        // 8,847,360
  float* h     = xcl + NX;                 // 18,399,744
  float* w1p   = h + NH;                   // 81*1*64   = 5,184
  float* w2p   = w1p + 81 * 1 * 64;        // 3*81*3*64 = 46,656
  float* parts = w2p + 3 * 81 * 3 * 64;    // 512*24    = 12,288
  float* ss    = parts + NBLK_STATS * 24;  // 24

  // 1) channel-last transpose of x
  transpose_cl<<<(NX + 255) / 256, 256, 0, stream>>>(x, xcl);

  // 2) weight fragment prep
  prep_weights<<<32, 256, 0, stream>>>(Wc1, w1p, CMID, CIN1, 1, 1);
  prep_weights<<<192, 256, 0, stream>>>(Wc2, w2p, COUT2, CMID, 3, 3);

  // 3) conv1 + bias + ReLU -> h (channel-last)
  conv4d_wmma<CIN1, CMID, 1, true, T0, D0, H0, W0, T1, D1, H1, W1d>
      <<<dim3(2048, 1), 256, 0, stream>>>(xcl, w1p, b1, h);

  // 4) batchnorm statistics (deterministic two-level reduction)
  stats_kernel<<<NBLK_STATS, 384, 0, stream>>>(h, parts);
  finalize_stats<<<1, 32, 0, stream>>>(parts, gamma, beta, ss);

  // 5) in-place normalize
  norm_kernel<<<2048, 384, 0, stream>>>(h, ss);

  // 6) conv2 + bias + ReLU -> d_out in reference [B,Cout,T',D',H',W'] order
  conv4d_wmma<CMID, COUT2, 3, false, T1, D1, H1, W1d, T2, D2, H2, W2d>
      <<<dim3(2048, 3), 256, 0, stream>>>(h, w2p, b2, out);
}